// EnhancedGraphLayer_52871047414155
// MI455X (gfx1250) — compile-verified
//
#include <hip/hip_runtime.h>

// EnhancedGraphLayer fused pipeline for MI455X (gfx1250, wave32, WMMA + TDM).
// bs=8, n=256, nin=64, ein=32, d=64. fp32 throughout (memory-bound; fp32 WMMA
// V_WMMA_F32_16X16X4_F32 keeps full precision at no bandwidth cost).
// E[b,n] tiles (32KB) are DMA'd into LDS by the Tensor Data Mover with
// hardware row padding (stride 33 dwords) for bank-conflict-free access.

#define BS  8
#define NN  256
#define NIN 64
#define EIN 32
#define DD  64
#define LN_EPS 1e-5f

typedef __attribute__((ext_vector_type(2))) float v2f;
typedef __attribute__((ext_vector_type(8))) float v8f;
typedef __attribute__((ext_vector_type(4))) unsigned int u32x4;
typedef __attribute__((ext_vector_type(8))) int i32x8;
typedef __attribute__((ext_vector_type(4))) int i32x4;

// ---------------------------------------------------------------------------
// TDM: DMA one E tile (256 rows x 32 f32, contiguous) from global into LDS,
// padding +1 DWORD after every 32 DWORDs => LDS row stride 33 floats.
// D# bitfields per CDNA5 ISA ch.8 (group0 128b, group1 256b, groups 2/3 zero
// for a 2D tile). Flat shared-aperture pointers carry the LDS byte offset in
// their low 32 bits (ISA 10.2 aperture mapping).
__device__ __forceinline__ void tdm_load_E_tile(const float* gsrc, float* lds_dst)
{
  const unsigned lds_off = (unsigned)(size_t)lds_dst;          // LDS byte offset
  const unsigned long long ga = (unsigned long long)(size_t)gsrc;

  u32x4 g0;
  g0[0] = 1u;                                   // count=1 (valid user descriptor)
  g0[1] = lds_off;                              // lds_addr
  g0[2] = (unsigned)(ga & 0xffffffffu);         // global_addr[31:0]
  g0[3] = (unsigned)((ga >> 32) & 0x01ffffffu)  // global_addr[56:32]
        | 0x80000000u;                          // type=2 (bits 127:126 = 10b)

  i32x8 g1;
  g1[0] = (2 << 16)       // data_size = 4 bytes
        | (1 << 20)       // pad_enable
        | (4 << 22);      // pad_interval code 4 -> pad after 32 DWORDs
                          // pad_amount = 0 -> 1 DWORD  => stride 33 floats
  g1[1] = (EIN << 16);    // tensor_dim0 = 32  (bits 79:48)
  g1[2] = (NN  << 16);    // tensor_dim1 = 256 (bits 111:80)
  g1[3] = (EIN << 16);    // tile_dim0   = 32  (bits 127:112)
  g1[4] = NN;             // tile_dim1   = 256 (bits 143:128), tile_dim2 = 0
  g1[5] = EIN;            // tensor_dim0_stride = 32 (bits 207:160)
  g1[6] = 0;
  g1[7] = 0;              // tensor_dim1_stride = 0 (unused for 2D)

  const i32x4 gz4 = {};
#if __clang_major__ >= 23
  const i32x8 gz8 = {};
  __builtin_amdgcn_tensor_load_to_lds(g0, g1, gz4, gz4, gz8, 0);
#else
  __builtin_amdgcn_tensor_load_to_lds(g0, g1, gz4, gz4, 0);
#endif
}

// ---------------------------------------------------------------------------
// One 16x16 f32 output tile of (A[M x EIN] @ B[EIN x DD]) from LDS, K=32 via
// eight chained V_WMMA_F32_16X16X4_F32.
// ISA layouts (CDNA5 7.12.2):
//   A 16x4 : lanes 0-15 -> M=lane, K={0,1} in v[0],v[1]; lanes 16-31 -> K={2,3}
//   B 4x16 : mirrored (row k striped across lanes within a VGPR)
//   C/D    : vgpr r -> (M=r, N=lane) for lanes 0-15, (M=r+8, N=lane-16) for 16-31
__device__ __forceinline__ void wmma_ep_tile(
    const float* sA, int strA, const float* sB, int strB,
    float* sOut, int strOut, int m0, int n0, int lane)
{
  const int mr = lane & 15;   // row / col within tile
  const int kh = lane >> 4;   // which K-half this lane holds
  v8f acc = {};
#pragma unroll
  for (int kb = 0; kb < EIN / 4; ++kb) {
    const int k = kb * 4 + kh * 2;
    v2f a, b;
    a.x = sA[(m0 + mr) * strA + k];
    a.y = sA[(m0 + mr) * strA + k + 1];
    b.x = sB[(k)     * strB + n0 + mr];
    b.y = sB[(k + 1) * strB + n0 + mr];
    acc = __builtin_amdgcn_wmma_f32_16x16x4_f32(
        /*neg_a=*/false, a, /*neg_b=*/false, b,
        /*c_mod=*/(short)0, acc, /*reuse_a=*/false, /*reuse_b=*/false);
  }
#pragma unroll
  for (int r = 0; r < 8; ++r)
    sOut[(m0 + r + kh * 8) * strOut + n0 + mr] = acc[r];
}

// ---------------------------------------------------------------------------
// k1: per-batch prep. local = X@W3 ; s_i = local@a1 ; s_j = local@a2 ;
//     gmean = mean_n(X) @ W5   (mean commutes with the matmul).
// Dynamic LDS: sX[256*64] + sL[256*65]
__global__ __launch_bounds__(256)
void egl_k1_prep(const float* __restrict__ X, const float* __restrict__ W3,
                 const float* __restrict__ W5, const float* __restrict__ a_w,
                 float* __restrict__ local_out, float* __restrict__ si,
                 float* __restrict__ sj, float* __restrict__ gmean)
{
  extern __shared__ float smem1[];
  float* sX = smem1;                    // NN*NIN
  float* sL = sX + NN * NIN;            // NN*(DD+1)  (padded rows)
  __shared__ float sxm[NIN];

  const int b = blockIdx.x;
  const int t = threadIdx.x;
  const float* Xb = X + (size_t)b * NN * NIN;

  for (int i = t; i < NN * NIN; i += 256) sX[i] = Xb[i];
  __syncthreads();

  // local = X @ W3 (thread = (dcol, row-group))
  const int dcol = t & 63, grp = t >> 6;
  for (int n = grp; n < NN; n += 4) {
    float acc = 0.f;
    for (int k = 0; k < NIN; ++k) acc += sX[n * NIN + k] * W3[k * DD + dcol];
    sL[n * (DD + 1) + dcol] = acc;
    local_out[((size_t)b * NN + n) * DD + dcol] = acc;
  }

  // column means of X
  if (t < NIN) {
    float s = 0.f;
    for (int n = 0; n < NN; ++n) s += sX[n * NIN + t];
    sxm[t] = s * (1.f / NN);
  }
  __syncthreads();

  if (t < DD) {
    float g = 0.f;
    for (int k = 0; k < NIN; ++k) g += sxm[k] * W5[k * DD + t];
    gmean[b * DD + t] = g;
  }

  // s_i, s_j (thread = node row)
  {
    float a1s = 0.f, a2s = 0.f;
    for (int dd2 = 0; dd2 < DD; ++dd2) {
      const float lv = sL[t * (DD + 1) + dd2];
      a1s += lv * a_w[dd2];
      a2s += lv * a_w[DD + dd2];
    }
    si[b * NN + t] = a1s;
    sj[b * NN + t] = a2s;
  }
}

// ---------------------------------------------------------------------------
// k2: per-(b,n). Ep = E[b,n] @ W4 in LDS (WMMA), s_e, softmax over m,
//     local_info, fused2 = (local_info*gmean)@W6, X_next = relu(LN(+local)).
// Dynamic LDS: sE[256*33] + sW4[32*65] + sEp[256*65]
__global__ __launch_bounds__(256)
void egl_k2_attn(const float* __restrict__ E, const float* __restrict__ W4,
                 const float* __restrict__ a_w, const float* __restrict__ local_in,
                 const float* __restrict__ si, const float* __restrict__ sj,
                 const float* __restrict__ gmean, const float* __restrict__ W6,
                 const float* __restrict__ gn, const float* __restrict__ bn,
                 float* __restrict__ Xnext, float* __restrict__ alpha_out)
{
  extern __shared__ float smem2[];
  float* sE  = smem2;                       // NN*(EIN+1)  (TDM pads rows)
  float* sW4 = sE + NN * (EIN + 1);         // EIN*(DD+1)
  float* sEp = sW4 + EIN * (DD + 1);        // NN*(DD+1)
  __shared__ float sVal[NN];                // alpha
  __shared__ float sRed[NN];                // reductions / y row
  __shared__ float sH[DD];                  // local_info * gmean
  __shared__ float sPart[4 * DD];           // partial sums

  const int n = blockIdx.x, b = blockIdx.y, t = threadIdx.x;
  const int lane = t & 31, wave = t >> 5;
  const size_t ebase = ((size_t)(b * NN + n)) * NN * EIN;

  // Async DMA of the 32KB E tile into LDS (wave 0 issues one descriptor).
  if (wave == 0) tdm_load_E_tile(E + ebase, sE);

  // Stage W4 (8KB) while the TDM runs.
  for (int i = t; i < EIN * DD; i += 256)
    sW4[(i >> 6) * (DD + 1) + (i & 63)] = W4[i];

  __builtin_amdgcn_s_wait_tensorcnt(0);
  __syncthreads();

  // Ep (256x64) via WMMA: 8 waves x {2 M-strips x 4 N-tiles}
#pragma unroll
  for (int s = 0; s < 2; ++s) {
    const int m0 = (wave + s * 8) * 16;
#pragma unroll
    for (int nt = 0; nt < 4; ++nt)
      wmma_ep_tile(sE, EIN + 1, sW4, DD + 1, sEp, DD + 1, m0, nt * 16, lane);
  }
  __syncthreads();

  // attention logits (thread = m) and softmax over m
  float se = 0.f;
  for (int dd2 = 0; dd2 < DD; ++dd2) se += sEp[t * (DD + 1) + dd2] * a_w[2 * DD + dd2];
  const float ev = (si[b * NN + n] + sj[b * NN + t] + se) * 0.125f; // 1/sqrt(64)
  sRed[t] = ev;
  __syncthreads();
  for (int off = 128; off > 0; off >>= 1) {
    if (t < off) sRed[t] = fmaxf(sRed[t], sRed[t + off]);
    __syncthreads();
  }
  const float emax = sRed[0];
  __syncthreads();
  const float p = __expf(ev - emax);
  sRed[t] = p;
  __syncthreads();
  for (int off = 128; off > 0; off >>= 1) {
    if (t < off) sRed[t] += sRed[t + off];
    __syncthreads();
  }
  const float psum = sRed[0];
  __syncthreads();
  const float alpha = p / psum;
  sVal[t] = alpha;
  alpha_out[((size_t)(b * NN + n)) * NN + t] = alpha;
  __syncthreads();

  // local_info[d] = sum_m alpha[m] * (local[b,m,d] + Ep[m,d])
  const int dcol = t & 63, grp = t >> 6;
  float li = 0.f;
  for (int m = grp; m < NN; m += 4) {
    const float a = sVal[m];
    li += a * (local_in[((size_t)b * NN + m) * DD + dcol] + sEp[m * (DD + 1) + dcol]);
  }
  sPart[grp * DD + dcol] = li;
  __syncthreads();
  if (t < DD) {
    const float v = sPart[t] + sPart[DD + t] + sPart[2 * DD + t] + sPart[3 * DD + t];
    sH[t] = v * gmean[b * DD + t];
  }
  __syncthreads();

  // fused2 + residual, then layernorm+relu across the 64-wide row
  if (t < DD) {
    float f = 0.f;
    for (int k = 0; k < DD; ++k) f += sH[k] * W6[k * DD + t];
    sRed[t] = f + local_in[((size_t)b * NN + n) * DD + t];
  }
  __syncthreads();
  if (t < DD) {
    float mean = 0.f;
    for (int k = 0; k < DD; ++k) mean += sRed[k];
    mean *= (1.f / DD);
    float var = 0.f;
    for (int k = 0; k < DD; ++k) { const float dv = sRed[k] - mean; var += dv * dv; }
    var *= (1.f / DD);
    const float xv = (sRed[t] - mean) * rsqrtf(var + LN_EPS) * gn[t] + bn[t];
    Xnext[((size_t)b * NN + n) * DD + t] = fmaxf(xv, 0.f);
  }
}

// ---------------------------------------------------------------------------
// k3: per-(b,n). Recompute Ep (WMMA; re-reading E's 67MB beats spilling a
//     268MB Ep round-trip), nis = alpha @ X_next, E_next = relu(LN(Ep + nis)).
// Dynamic LDS: sE[256*33] + sW4[32*65] + sEp[256*65]
__global__ __launch_bounds__(256)
void egl_k3_edge(const float* __restrict__ E, const float* __restrict__ W4,
                 const float* __restrict__ alpha_in, const float* __restrict__ Xnext,
                 const float* __restrict__ ge, const float* __restrict__ be,
                 float* __restrict__ Enext)
{
  extern __shared__ float smem3[];
  float* sE  = smem3;
  float* sW4 = sE + NN * (EIN + 1);
  float* sEp = sW4 + EIN * (DD + 1);
  __shared__ float sAl[NN];
  __shared__ float sNis[DD];
  __shared__ float sPart[4 * DD];
  __shared__ float sMean[NN];
  __shared__ float sRstd[NN];

  const int n = blockIdx.x, b = blockIdx.y, t = threadIdx.x;
  const int lane = t & 31, wave = t >> 5;
  const size_t ebase = ((size_t)(b * NN + n)) * NN * EIN;

  if (wave == 0) tdm_load_E_tile(E + ebase, sE);

  sAl[t] = alpha_in[((size_t)(b * NN + n)) * NN + t];
  for (int i = t; i < EIN * DD; i += 256)
    sW4[(i >> 6) * (DD + 1) + (i & 63)] = W4[i];

  __builtin_amdgcn_s_wait_tensorcnt(0);
  __syncthreads();

  // node_interact_sum partials (alpha @ X_next), overlapped with WMMA issue
  const int dcol = t & 63, grp = t >> 6;
  float pn = 0.f;
  for (int m = grp; m < NN; m += 4)
    pn += sAl[m] * Xnext[((size_t)b * NN + m) * DD + dcol];
  sPart[grp * DD + dcol] = pn;

#pragma unroll
  for (int s = 0; s < 2; ++s) {
    const int m0 = (wave + s * 8) * 16;
#pragma unroll
    for (int nt = 0; nt < 4; ++nt)
      wmma_ep_tile(sE, EIN + 1, sW4, DD + 1, sEp, DD + 1, m0, nt * 16, lane);
  }
  __syncthreads();
  if (t < DD) sNis[t] = sPart[t] + sPart[DD + t] + sPart[2 * DD + t] + sPart[3 * DD + t];
  __syncthreads();

  // per-edge-row layernorm stats (thread = m; padded stride -> conflict-free)
  {
    float s1 = 0.f, s2 = 0.f;
    for (int dd2 = 0; dd2 < DD; ++dd2) {
      const float v = sEp[t * (DD + 1) + dd2] + sNis[dd2];
      s1 += v; s2 += v * v;
    }
    const float mean = s1 * (1.f / DD);
    const float var = s2 * (1.f / DD) - mean * mean;
    sMean[t] = mean;
    sRstd[t] = rsqrtf(fmaxf(var, 0.f) + LN_EPS);
  }
  __syncthreads();

  // coalesced store of the normalized tile
  float* outb = Enext + ((size_t)(b * NN + n)) * NN * DD;
  for (int i = t; i < NN * DD; i += 256) {
    const int m = i >> 6, dd2 = i & 63;
    const float v = sEp[m * (DD + 1) + dd2] + sNis[dd2];
    const float xv = (v - sMean[m]) * sRstd[m] * ge[dd2] + be[dd2];
    outb[i] = fmaxf(xv, 0.f);
  }
}

// ---------------------------------------------------------------------------
extern "C" void kernel_launch(void* const* d_in, const int* in_sizes, int n_in,
                              void* d_out, int out_size, void* d_ws, size_t ws_size,
                              hipStream_t stream)
{
  (void)in_sizes; (void)n_in; (void)out_size; (void)ws_size;
  const float* X   = (const float*)d_in[0];
  const float* E   = (const float*)d_in[1];
  const float* W3  = (const float*)d_in[2];
  const float* W4  = (const float*)d_in[3];
  const float* W5  = (const float*)d_in[4];
  const float* W6  = (const float*)d_in[5];
  const float* a_w = (const float*)d_in[6];
  const float* gn  = (const float*)d_in[7];
  const float* bn  = (const float*)d_in[8];
  const float* ge  = (const float*)d_in[9];
  const float* be  = (const float*)d_in[10];

  float* Xnext = (float*)d_out;                 // BS*NN*DD
  float* Enext = Xnext + (size_t)BS * NN * DD;  // BS*NN*NN*DD

  float* ws      = (float*)d_ws;
  float* w_local = ws;                              // BS*NN*DD
  float* w_si    = w_local + (size_t)BS * NN * DD;  // BS*NN
  float* w_sj    = w_si + BS * NN;                  // BS*NN
  float* w_gm    = w_sj + BS * NN;                  // BS*DD
  float* w_al    = w_gm + BS * DD;                  // BS*NN*NN

  const size_t lds1  = (size_t)(NN * NIN + NN * (DD + 1)) * sizeof(float);
  const size_t lds23 = (size_t)(NN * (EIN + 1) + EIN * (DD + 1) + NN * (DD + 1)) * sizeof(float);

  egl_k1_prep<<<dim3(BS), 256, lds1, stream>>>(X, W3, W5, a_w, w_local, w_si, w_sj, w_gm);
  egl_k2_attn<<<dim3(NN, BS), 256, lds23, stream>>>(E, W4, a_w, w_local, w_si, w_sj,
                                                    w_gm, W6, gn, bn, Xnext, w_al);
  egl_k3_edge<<<dim3(NN, BS), 256, lds23, stream>>>(E, W4, w_al, Xnext, ge, be, Enext);
}